// SpMergeAttentionLayer_25262997635460
// MI455X (gfx1250) — compile-verified
//
#include <hip/hip_runtime.h>

// ---------------------------------------------------------------------------
// fp32 WMMA GEMM (V_WMMA_F32_16X16X4_F32), plus graph-attention pipeline.
// ---------------------------------------------------------------------------

typedef float v2f __attribute__((ext_vector_type(2)));
typedef float v8f __attribute__((ext_vector_type(8)));

#define TM 64      // block tile rows
#define TNB 64     // block tile cols
#define TKC 32     // K chunk
#define A_PAD 36   // LDS row stride for A tile: conflict-free frag reads, 16B rows
#define B_PAD 34   // LDS row stride for Bt tile: even (8B-aligned b64 frag reads)

__device__ __forceinline__ void atomicAddF32(float* p, float v)
{
#if defined(__HIP_DEVICE_COMPILE__)
    unsafeAtomicAdd(p, v);   // native global_atomic_add_f32 on CDNA
#else
    atomicAdd(p, v);
#endif
}

// C[M,N] = A[M,K] @ B[K,N] (+ bias[n]), fp32 row-major. N mult of 64, K mult of 32.
__global__ __launch_bounds__(256) void gemm_f32_wmma(
    const float* __restrict__ A, const float* __restrict__ B,
    const float* __restrict__ bias, float* __restrict__ C,
    int M, int K, int N)
{
    __shared__ float As[TM * A_PAD];     // A tile, row-major [64][36]
    __shared__ float Bt[TNB * B_PAD];    // B tile transposed [n][k] = [64][34]

    const int tid  = threadIdx.x;
    const int lane = tid & 31;
    const int wave = tid >> 5;      // 0..7
    const int wr   = wave >> 1;     // 0..3  (16-row wave tile)
    const int wc   = wave & 1;      // 0..1  (32-col wave tile = two 16-col WMMA tiles)
    const int row0 = blockIdx.y * TM;
    const int col0 = blockIdx.x * TNB;

    const int m  = lane & 15;       // A row / B col / C col within 16x16 tile
    const int kh = lane >> 4;       // lane half: K pair select (k+0,1 vs k+2,3)
    const int n0 = wc * 32;         // first 16-col sub-tile base (local)

    v8f acc0 = {};
    v8f acc1 = {};

    for (int k0 = 0; k0 < K; k0 += TKC) {
        // --- stage A tile: 64 x 32 floats = 512 float4, 2 per thread, b128 stores
        #pragma unroll
        for (int it = 0; it < 2; ++it) {
            int fi = tid + it * 256;        // 0..511
            int r  = fi >> 3;               // 0..63
            int c4 = fi & 7;                // float4 column
            int gr = row0 + r;
            float4 v = make_float4(0.f, 0.f, 0.f, 0.f);
            if (gr < M)
                v = *(const float4*)(A + (size_t)gr * K + k0 + c4 * 4);
            *(float4*)&As[r * A_PAD + c4 * 4] = v;   // row stride 144B (16B mult)
        }
        // --- stage B tile transposed: 32(k) x 64(n) floats, scatter to Bt[n][k]
        #pragma unroll
        for (int it = 0; it < 2; ++it) {
            int fi = tid + it * 256;        // 0..511
            int r  = fi >> 4;               // k row 0..31
            int c4 = fi & 15;               // float4 n-column
            float4 v = *(const float4*)(B + (size_t)(k0 + r) * N + col0 + c4 * 4);
            Bt[(c4 * 4 + 0) * B_PAD + r] = v.x;
            Bt[(c4 * 4 + 1) * B_PAD + r] = v.y;
            Bt[(c4 * 4 + 2) * B_PAD + r] = v.z;
            Bt[(c4 * 4 + 3) * B_PAD + r] = v.w;
        }
        __syncthreads();

        // --- 16 x V_WMMA_F32_16X16X4_F32 over the 32-wide K chunk (2 acc chains)
        const int arow = (wr * 16 + m) * A_PAD;
        const int brow0 = (n0 + m) * B_PAD;
        const int brow1 = (n0 + 16 + m) * B_PAD;
        #pragma unroll
        for (int kk = 0; kk < TKC; kk += 4) {
            int ko = kk + 2 * kh;
            // A 16x4 frag: lanes 0-15 hold (K=kk,kk+1); lanes 16-31 (K=kk+2,kk+3)
            v2f a  = *(const v2f*)&As[arow + ko];
            // B 4x16 frag mirrored: reg = k&1, lane-half = k>>1 (Bt is [n][k])
            v2f b0 = *(const v2f*)&Bt[brow0 + ko];
            v2f b1 = *(const v2f*)&Bt[brow1 + ko];
            acc0 = __builtin_amdgcn_wmma_f32_16x16x4_f32(
                false, a, false, b0, (short)0, acc0, false, false);
            acc1 = __builtin_amdgcn_wmma_f32_16x16x4_f32(
                false, a, false, b1, (short)0, acc1, false, false);
        }
        __syncthreads();
    }

    // --- epilogue: C/D layout reg r -> M = r + 8*(lane>=16), lane%16 -> N ---
    const int gc0 = col0 + n0 + m;
    const int gc1 = gc0 + 16;
    const float bv0 = bias ? bias[gc0] : 0.0f;
    const float bv1 = bias ? bias[gc1] : 0.0f;
    #pragma unroll
    for (int r = 0; r < 8; ++r) {
        int gm = row0 + wr * 16 + r + 8 * kh;
        if (gm < M) {
            C[(size_t)gm * N + gc0] = acc0[r] + bv0;
            C[(size_t)gm * N + gc1] = acc1[r] + bv1;
        }
    }
}

// ---------------------------------------------------------------------------
// Wr[r] = att[r][0]*basis[0] + att[r][1]*basis[1]   (elementwise, 256*256 each)
__global__ void make_wr(const float* __restrict__ att,
                        const float* __restrict__ basis,
                        float* __restrict__ wr, int n /* = 256*256 */)
{
    int t = blockIdx.x * blockDim.x + threadIdx.x;
    if (t >= n) return;
    float b0 = basis[t], b1 = basis[n + t];
    wr[t]     = att[0] * b0 + att[1] * b1;
    wr[n + t] = att[2] * b0 + att[3] * b1;
}

// ---------------------------------------------------------------------------
// Per-node score scalars: s[i]=h[i].w_src, d[i]=h[i].w_dst (both h_pos & h_neg)
__global__ __launch_bounds__(256) void node_scores(
    const float* __restrict__ hp, const float* __restrict__ hn,
    const float* __restrict__ mapping,   // [0:256]=w_src, [256:512]=w_dst
    float* __restrict__ sp, float* __restrict__ dp,
    float* __restrict__ sn, float* __restrict__ dn, int N)
{
    int wave = blockIdx.x * (blockDim.x >> 5) + (threadIdx.x >> 5);
    int lane = threadIdx.x & 31;
    if (wave >= N) return;
    const float* rp = hp + (size_t)wave * 256;
    const float* rn = hn + (size_t)wave * 256;
    float asp = 0.f, adp = 0.f, asn = 0.f, adn = 0.f;
    #pragma unroll
    for (int j = lane; j < 256; j += 32) {
        float ws = mapping[j], wd = mapping[256 + j];
        float vp = rp[j],      vn = rn[j];
        asp += vp * ws; adp += vp * wd;
        asn += vn * ws; adn += vn * wd;
    }
    #pragma unroll
    for (int off = 16; off > 0; off >>= 1) {
        asp += __shfl_down(asp, off, 32);
        adp += __shfl_down(adp, off, 32);
        asn += __shfl_down(asn, off, 32);
        adn += __shfl_down(adn, off, 32);
    }
    if (lane == 0) { sp[wave] = asp; dp[wave] = adp; sn[wave] = asn; dn[wave] = adn; }
}

// ---------------------------------------------------------------------------
// Per-edge raw coefficient (sigmoid(leakyrelu)) + degree counting
__global__ void edge_coeff(const int* __restrict__ adjp, const int* __restrict__ adjn,
                           const float* __restrict__ sp, const float* __restrict__ dp,
                           const float* __restrict__ sn, const float* __restrict__ dn,
                           float* __restrict__ coeff, float* __restrict__ degcnt, int E)
{
    int t = blockIdx.x * blockDim.x + threadIdx.x;
    if (t >= 2 * E) return;
    int src, dst; float e;
    if (t < E) { src = adjp[t];      dst = adjp[E + t];      e = sp[src] + dp[dst]; }
    else       { int i = t - E;
                 src = adjn[i];      dst = adjn[E + i];      e = sn[src] + dn[dst]; }
    float l = (e > 0.f) ? e : 0.2f * e;
    coeff[t] = 1.0f / (1.0f + __expf(-l));
    atomicAddF32(&degcnt[src], 1.0f);
}

__global__ void deg_sqrt(float* __restrict__ deg, int N)
{
    int t = blockIdx.x * blockDim.x + threadIdx.x;
    if (t < N) deg[t] = sqrtf(deg[t]);
}

// ---------------------------------------------------------------------------
// Scatter aggregation: one wave per edge, 8 floats/lane, native f32 atomics
__global__ __launch_bounds__(256) void aggregate(
    const int* __restrict__ adjp, const int* __restrict__ adjn,
    const float* __restrict__ hp, const float* __restrict__ hn,
    const float* __restrict__ coeff, const float* __restrict__ deg,
    float* __restrict__ aggp, float* __restrict__ aggn, int E)
{
    int wave = blockIdx.x * (blockDim.x >> 5) + (threadIdx.x >> 5);
    int lane = threadIdx.x & 31;
    if (wave >= 2 * E) return;
    int src, dst; const float* h; float* agg;
    if (wave < E) { src = adjp[wave]; dst = adjp[E + wave]; h = hp; agg = aggp; }
    else          { int i = wave - E;
                    src = adjn[i];    dst = adjn[E + i];    h = hn; agg = aggn; }
    float c = coeff[wave] / (deg[src] * deg[dst]);
    const float* hr = h   + (size_t)dst * 256;
    float*       ar = agg + (size_t)src * 256;
    #pragma unroll
    for (int j = 0; j < 8; ++j) {
        int col = lane + j * 32;
        atomicAddF32(&ar[col], c * hr[col]);
    }
}

// ---------------------------------------------------------------------------
__global__ void finalize(const float* __restrict__ aggp, const float* __restrict__ aggn,
                         const float* __restrict__ bias, float* __restrict__ out, int total)
{
    int t = blockIdx.x * blockDim.x + threadIdx.x;
    if (t < total) out[t] = aggp[t] - aggn[t] + bias[t & 255];
}

__global__ void fill_zero(float* __restrict__ p, long n)
{
    long i = blockIdx.x * (long)blockDim.x + threadIdx.x;
    long s = (long)gridDim.x * blockDim.x;
    for (; i < n; i += s) p[i] = 0.f;
}

// ---------------------------------------------------------------------------
extern "C" void kernel_launch(void* const* d_in, const int* in_sizes, int n_in,
                              void* d_out, int out_size, void* d_ws, size_t ws_size,
                              hipStream_t stream)
{
    const float* X       = (const float*)d_in[0];   // (N,256)
    const int*   adjp    = (const int*)  d_in[1];   // (2,E)
    const int*   adjn    = (const int*)  d_in[2];   // (2,E)
    const float* basis   = (const float*)d_in[3];   // (2,256,256)
    const float* att     = (const float*)d_in[4];   // (2,2)
    const float* mapping = (const float*)d_in[5];   // (1,512)
    const float* bias    = (const float*)d_in[6];   // (1,256)
    const float* th1     = (const float*)d_in[7];   // (256,256)
    const float* b1      = (const float*)d_in[8];   // (256,)
    const float* th2     = (const float*)d_in[9];   // (256,256)
    const float* b2      = (const float*)d_in[10];  // (256,)

    const int D  = 256;
    const int N  = in_sizes[0] / D;        // 50000
    const int E  = in_sizes[1] / 2;        // 400000
    const int WW = D * D;                  // 65536

    // d_out sections: [output | h_agg_pos | h_agg_neg]
    float* out0 = (float*)d_out;
    float* aggp = out0 + (size_t)N * D;
    float* aggn = aggp + (size_t)N * D;

    // workspace layout
    float* Wr     = (float*)d_ws;              // 2*WW
    float* Wc     = Wr + 2 * WW;               // 2*WW  (Wpos, Wneg combined weights)
    float* hp     = Wc + 2 * WW;               // N*D
    float* hn     = hp + (size_t)N * D;        // N*D
    float* sp     = hn + (size_t)N * D;        // N
    float* dp     = sp + N;                    // N
    float* sn     = dp + N;                    // N
    float* dn     = sn + N;                    // N
    float* coeff  = dn + N;                    // 2*E
    float* degcnt = coeff + 2 * (size_t)E;     // N

    // 0) zero accumulators (harness poisons once; we must re-zero every call)
    fill_zero<<<2048, 256, 0, stream>>>(aggp, 2L * N * D);
    fill_zero<<<64,   256, 0, stream>>>(degcnt, (long)N);

    // 1) Wr = att @ basis  (2 x 256 x 256)
    make_wr<<<(WW + 255) / 256, 256, 0, stream>>>(att, basis, Wr, WW);

    // 2) combined weights: Wpos = Wr0 @ theta1, Wneg = Wr1 @ theta2 (exact collapse)
    dim3 gw(D / TNB, (D + TM - 1) / TM);
    gemm_f32_wmma<<<gw, 256, 0, stream>>>(Wr,      th1, nullptr, Wc,      D, D, D);
    gemm_f32_wmma<<<gw, 256, 0, stream>>>(Wr + WW, th2, nullptr, Wc + WW, D, D, D);

    // 3) h_pos = X @ Wpos + b1 ; h_neg = X @ Wneg + b2   (big WMMA GEMMs)
    dim3 gh(D / TNB, (N + TM - 1) / TM);
    gemm_f32_wmma<<<gh, 256, 0, stream>>>(X, Wc,      b1, hp, N, D, D);
    gemm_f32_wmma<<<gh, 256, 0, stream>>>(X, Wc + WW, b2, hn, N, D, D);

    // 4) per-node score scalars (turns per-edge 1KB gathers into 2 scalar gathers)
    node_scores<<<(N + 7) / 8, 256, 0, stream>>>(hp, hn, mapping, sp, dp, sn, dn, N);

    // 5) per-edge coeff + degree counts
    edge_coeff<<<(2 * E + 255) / 256, 256, 0, stream>>>(adjp, adjn, sp, dp, sn, dn,
                                                        coeff, degcnt, E);
    deg_sqrt<<<(N + 255) / 256, 256, 0, stream>>>(degcnt, N);

    // 6) normalized scatter aggregation (native global_atomic_add_f32)
    aggregate<<<(2 * E + 7) / 8, 256, 0, stream>>>(adjp, adjn, hp, hn, coeff, degcnt,
                                                   aggp, aggn, E);

    // 7) output = h_agg_pos - h_agg_neg + bias
    finalize<<<((N * D) + 255) / 256, 256, 0, stream>>>(aggp, aggn, bias, out0, N * D);
}